// Net_60567628808344
// MI455X (gfx1250) — compile-verified
//
#include <hip/hip_runtime.h>
#include <hip/hip_bf16.h>

// DMP rollout: out[row][t] = y_t for 131072 rows, 301 steps.
// F = (w * (goal-y0)) @ Phi  computed with V_WMMA_F32_16X16X4_F32,
// recurrence + coalesced stores via per-wave LDS tiles.
// Divisions strength-reduced to constants so the per-step chain is pure FMA.

typedef float v2f __attribute__((ext_vector_type(2)));
typedef float v8f __attribute__((ext_vector_type(8)));

#define Nc      25
#define NP2     27          // N+2
#define PDIM    54          // DOF*(N+2)
#define TSTEPS  301
#define TPAD    304         // 19 chunks of 16
#define NCHUNK  19
#define KPAD    28          // 7 chunks of K=4
#define TAUc    3.0f
#define DTc     0.01f
#define A_Xc    2.0f
#define A_Zc    48.0f
#define B_Zc    12.0f
#define AZBZ    (A_Zc * B_Zc)            // 576
#define KDT     (DTc / TAUc)             // 0.0033333333

__global__ __launch_bounds__(256) void dmp_wmma_kernel(
    const float* __restrict__ x,
    const float* __restrict__ cvec,
    const float* __restrict__ sigma2,
    const float* __restrict__ scale,
    float* __restrict__ out)
{
    __shared__ float s_cx[TPAD];
    __shared__ float s_nis[Nc];               // -0.5 / sigma2[k]
    __shared__ float s_cv[Nc];                // cvec[k]
    __shared__ float s_phi[KPAD * TPAD];      // [k][t], rows 25..27 zero
    __shared__ float s_tile[8][32 * 17];      // per-wave 32 rows x 16 steps (stride 17)

    const int tid  = threadIdx.x;
    const int lane = tid & 31;
    const int wave = tid >> 5;
    const int half = lane >> 4;               // 0: lanes 0-15, 1: lanes 16-31
    const int lm   = lane & 15;
    const int rowbase = blockIdx.x * 256 + wave * 32;

    // ---- build cx table exactly like the reference recurrence ----
    if (tid == 0) {
        float cv = 1.0f;
        for (int t = 0; t < TPAD; ++t) {
            cv = cv + ((-A_Xc * cv / TAUc) * DTc);   // cx update happens BEFORE psi
            s_cx[t] = cv;
        }
    }
    if (tid < Nc) {
        s_nis[tid] = -0.5f / sigma2[tid];
        s_cv[tid]  = cvec[tid];
    }
    __syncthreads();

    // ---- build normalized phi table: phi[k][t] = psi/sum(psi) * cx ----
    for (int t = tid; t < TPAD; t += 256) {
        float cxv = s_cx[t];
        float ps[Nc];
        float sum = 0.0f;
        #pragma unroll
        for (int k = 0; k < Nc; ++k) {
            float d = cxv - s_cv[k];
            float p = expf(d * d * s_nis[k]);
            ps[k] = p;
            sum += p;
        }
        float fac = cxv / sum;
        #pragma unroll
        for (int k = 0; k < Nc; ++k) s_phi[k * TPAD + t] = ps[k] * fac;
        s_phi[25 * TPAD + t] = 0.0f;
        s_phi[26 * TPAD + t] = 0.0f;
        s_phi[27 * TPAD + t] = 0.0f;
    }
    __syncthreads();

    // ---- load A operands: two 16-row tiles per wave, gain folded in ----
    // f32 A 16x4 layout: lanes 0-15 = M, vgpr0 = K0 (lanes<16) / K2 (lanes>=16),
    // vgpr1 = K1 / K3.  aT[tile][2c+j] is the j-th element of chunk c's v2f.
    float aT[2][14];
    #pragma unroll
    for (int tile = 0; tile < 2; ++tile) {
        int grow  = rowbase + tile * 16 + lm;
        int b_    = grow >> 1;
        int d_    = grow & 1;
        int pbase = b_ * PDIM + d_ * NP2;
        int soff  = d_ * NP2;
        float y0v = x[pbase]     * scale[soff];
        float glv = x[pbase + 1] * scale[soff + 1];
        float gf  = glv - y0v;
        #pragma unroll
        for (int c7 = 0; c7 < 7; ++c7) {
            int k0 = 4 * c7 + half * 2;
            float w0 = (k0     < Nc) ? x[pbase + 2 + k0]     * scale[soff + 2 + k0]     : 0.0f;
            float w1 = (k0 + 1 < Nc) ? x[pbase + 2 + k0 + 1] * scale[soff + 2 + k0 + 1] : 0.0f;
            aT[tile][2 * c7]     = w0 * gf;
            aT[tile][2 * c7 + 1] = w1 * gf;
        }
    }

    // ---- per-lane recurrence state: lane l owns row rowbase+l ----
    float y, z, gl;
    {
        int grow  = rowbase + lane;
        int b_    = grow >> 1;
        int d_    = grow & 1;
        int pbase = b_ * PDIM + d_ * NP2;
        int soff  = d_ * NP2;
        y  = x[pbase]     * scale[soff];
        gl = x[pbase + 1] * scale[soff + 1];
        z  = 0.0f;
    }

    float* Yt = &s_tile[wave][0];
    const int rloc = half * 8;                // D tile: lanes<16 hold M=0..7, else M=8..15

    for (int ch = 0; ch < NCHUNK; ++ch) {
        const int t0   = ch * 16;
        const int tcol = t0 + lm;

        v8f acc0 = {0.f,0.f,0.f,0.f,0.f,0.f,0.f,0.f};
        v8f acc1 = {0.f,0.f,0.f,0.f,0.f,0.f,0.f,0.f};

        #pragma unroll
        for (int c7 = 0; c7 < 7; ++c7) {
            int k0 = 4 * c7 + half * 2;
            v2f bv = { s_phi[k0 * TPAD + tcol], s_phi[(k0 + 1) * TPAD + tcol] };
            v2f a0 = { aT[0][2 * c7], aT[0][2 * c7 + 1] };
            v2f a1 = { aT[1][2 * c7], aT[1][2 * c7 + 1] };
            acc0 = __builtin_amdgcn_wmma_f32_16x16x4_f32(
                false, a0, false, bv, (short)0, acc0, false, false);
            acc1 = __builtin_amdgcn_wmma_f32_16x16x4_f32(
                false, a1, false, bv, (short)0, acc1, false, false);
        }

        // spill D tiles: F[row][step] -> per-wave LDS tile (row stride 17)
        #pragma unroll
        for (int j = 0; j < 8; ++j) {
            Yt[(rloc + j) * 17 + lm]        = acc0[j];
            Yt[(16 + rloc + j) * 17 + lm]   = acc1[j];
        }
        asm volatile("s_wait_dscnt 0x0" ::: "memory");   // intra-wave cross-lane handoff

        // lane l integrates its row for 16 steps, writes y back in place.
        // y += z*KDT ; z += (AZBZ*(gl-y_old) - A_Z*z_old + fx)*KDT   (pure FMA chain)
        float fxv[16];
        #pragma unroll
        for (int s = 0; s < 16; ++s) fxv[s] = Yt[lane * 17 + s];
        #pragma unroll
        for (int s = 0; s < 16; ++s) {
            float e  = gl - y;
            float dz = fmaf(AZBZ, e, fmaf(-A_Zc, z, fxv[s]));
            y = fmaf(z,  KDT, y);
            z = fmaf(dz, KDT, z);
            Yt[lane * 17 + s] = y;
        }
        asm volatile("s_wait_dscnt 0x0" ::: "memory");

        // coalesced drain: half-wave per row, 16 consecutive floats each
        if (tcol < TSTEPS) {
            #pragma unroll
            for (int i = 0; i < 16; ++i) {
                int r = 2 * i + half;
                out[(size_t)(rowbase + r) * TSTEPS + tcol] = Yt[r * 17 + lm];
            }
        }
        // next chunk's LDS stores are same-wave in-order vs. these loads: no wait needed
    }
}

extern "C" void kernel_launch(void* const* d_in, const int* in_sizes, int n_in,
                              void* d_out, int out_size, void* d_ws, size_t ws_size,
                              hipStream_t stream) {
    const float* x      = (const float*)d_in[0];
    const float* cvec   = (const float*)d_in[1];
    const float* sigma2 = (const float*)d_in[2];
    const float* scale  = (const float*)d_in[3];
    float* out = (float*)d_out;

    int batch  = in_sizes[0] / PDIM;   // 65536
    int rows   = batch * 2;            // 131072
    int blocks = rows / 256;           // 512 blocks x 256 threads (8 wave32, 32 rows/wave)

    dmp_wmma_kernel<<<blocks, 256, 0, stream>>>(x, cvec, sigma2, scale, out);
}